// KalmanFilter_57921928954599
// MI455X (gfx1250) — compile-verified
//
#include <hip/hip_runtime.h>

typedef __attribute__((ext_vector_type(2))) float v2f;
typedef __attribute__((ext_vector_type(8))) float v8f;

#define NTRK_BLK 256
#define MMEAS    128
#define FSTR     17   // 16 features + 1 pad (17 coprime with 64 banks)

__global__ __launch_bounds__(NTRK_BLK) void kalman_fused_kernel(
    const float* __restrict__ mean,   // [N,8]
    const float* __restrict__ cov,    // [N,8,8]
    const float* __restrict__ meas,   // [N,4]
    const float* __restrict__ gmeas,  // [M,4]
    const float* __restrict__ Fm,     // [8,8]
    const float* __restrict__ Hm,     // [4,8]
    const float* __restrict__ Qm,     // [8,8]
    const float* __restrict__ Rm,     // [4,4]
    float* __restrict__ out_mean,     // [N,8]
    float* __restrict__ out_cov,      // [N,8,8]
    float* __restrict__ out_maha,     // [N,M]
    int N, int M)
{
    __shared__ float sF[64];
    __shared__ float sH[32];
    __shared__ float sQ[64];
    __shared__ float sR[16];
    __shared__ float sTrk[NTRK_BLK][FSTR];   // per-track gating features
    __shared__ float sMeas[MMEAS][FSTR];     // per-measurement gating features

    const int tid = threadIdx.x;

    // ---- stage uniform model matrices ----
    if (tid < 64)                 sF[tid]       = Fm[tid];
    if (tid >= 64  && tid < 96)   sH[tid - 64]  = Hm[tid - 64];
    if (tid >= 96  && tid < 160)  sQ[tid - 96]  = Qm[tid - 96];
    if (tid >= 160 && tid < 176)  sR[tid - 160] = Rm[tid - 160];

    // ---- measurement feature vectors: q = [m_i*m_j (i<=j, 10), m (4), 1, 0] ----
    if (tid < M && tid < MMEAS) {
        float mm[4];
        #pragma unroll
        for (int i = 0; i < 4; ++i) mm[i] = gmeas[tid * 4 + i];
        int idx = 0;
        #pragma unroll
        for (int i = 0; i < 4; ++i)
            #pragma unroll
            for (int j = i; j < 4; ++j)
                sMeas[tid][idx++] = mm[i] * mm[j];
        #pragma unroll
        for (int i = 0; i < 4; ++i) sMeas[tid][10 + i] = mm[i];
        sMeas[tid][14] = 1.0f;
        sMeas[tid][15] = 0.0f;
    }
    __syncthreads();

    // =====================================================================
    // Phase 1: per-track Kalman predict -> project -> Cholesky -> update
    // =====================================================================
    const int track = blockIdx.x * NTRK_BLK + tid;
    if (track < N) {
        // load state
        float x[8];
        {
            const float4* mv = (const float4*)(mean + (size_t)track * 8);
            float4 a = mv[0], b = mv[1];
            x[0]=a.x; x[1]=a.y; x[2]=a.z; x[3]=a.w;
            x[4]=b.x; x[5]=b.y; x[6]=b.z; x[7]=b.w;
        }
        float P[8][8];
        {
            const float4* pv = (const float4*)(cov + (size_t)track * 64);
            #pragma unroll
            for (int i = 0; i < 8; ++i) {
                float4 a = pv[i*2+0], b = pv[i*2+1];
                P[i][0]=a.x; P[i][1]=a.y; P[i][2]=a.z; P[i][3]=a.w;
                P[i][4]=b.x; P[i][5]=b.y; P[i][6]=b.z; P[i][7]=b.w;
            }
        }
        float zdet[4];
        {
            const float4* zv = (const float4*)(meas + (size_t)track * 4);
            float4 a = zv[0];
            zdet[0]=a.x; zdet[1]=a.y; zdet[2]=a.z; zdet[3]=a.w;
        }

        // predict mean: pm = F x
        float pm[8];
        #pragma unroll
        for (int i = 0; i < 8; ++i) {
            float s = 0.f;
            #pragma unroll
            for (int j = 0; j < 8; ++j) s += sF[i*8+j] * x[j];
            pm[i] = s;
        }
        // PFt[j][l] = sum_k P[j][k] F[l][k]
        float PFt[8][8];
        #pragma unroll
        for (int j = 0; j < 8; ++j)
            #pragma unroll
            for (int l = 0; l < 8; ++l) {
                float s = 0.f;
                #pragma unroll
                for (int k = 0; k < 8; ++k) s += P[j][k] * sF[l*8+k];
                PFt[j][l] = s;
            }
        // PC = F PFt + Q
        float PC[8][8];
        #pragma unroll
        for (int i = 0; i < 8; ++i)
            #pragma unroll
            for (int l = 0; l < 8; ++l) {
                float s = sQ[i*8+l];
                #pragma unroll
                for (int j = 0; j < 8; ++j) s += sF[i*8+j] * PFt[j][l];
                PC[i][l] = s;
            }
        // projected mean: zm = H pm
        float zm[4];
        #pragma unroll
        for (int i = 0; i < 4; ++i) {
            float s = 0.f;
            #pragma unroll
            for (int j = 0; j < 8; ++j) s += sH[i*8+j] * pm[j];
            zm[i] = s;
        }
        // HP = H PC ; S = HP H^T + R
        float HP[4][8];
        #pragma unroll
        for (int i = 0; i < 4; ++i)
            #pragma unroll
            for (int k = 0; k < 8; ++k) {
                float s = 0.f;
                #pragma unroll
                for (int j = 0; j < 8; ++j) s += sH[i*8+j] * PC[j][k];
                HP[i][k] = s;
            }
        float S[4][4];
        #pragma unroll
        for (int i = 0; i < 4; ++i)
            #pragma unroll
            for (int l = 0; l < 4; ++l) {
                float s = sR[i*4+l];
                #pragma unroll
                for (int k = 0; k < 8; ++k) s += HP[i][k] * sH[l*8+k];
                S[i][l] = s;
            }
        // Cholesky S = L L^T (lower)
        float L[4][4];
        #pragma unroll
        for (int j = 0; j < 4; ++j) {
            float d = S[j][j];
            #pragma unroll
            for (int k = 0; k < 4; ++k) if (k < j) d -= L[j][k] * L[j][k];
            float lj = sqrtf(d);
            L[j][j] = lj;
            float rlj = 1.0f / lj;
            #pragma unroll
            for (int i = 0; i < 4; ++i) if (i > j) {
                float s = S[i][j];
                #pragma unroll
                for (int k = 0; k < 4; ++k) if (k < j) s -= L[i][k] * L[j][k];
                L[i][j] = s * rlj;
            }
        }
        // Bt[i][k] = sum_j H[i][j] PC[k][j]   (== (PC H^T)^T)
        float Bt[4][8];
        #pragma unroll
        for (int i = 0; i < 4; ++i)
            #pragma unroll
            for (int k = 0; k < 8; ++k) {
                float s = 0.f;
                #pragma unroll
                for (int j = 0; j < 8; ++j) s += sH[i*8+j] * PC[k][j];
                Bt[i][k] = s;
            }
        // forward solve L Y = Bt ; backward solve L^T X = Y
        float Y[4][8], X[4][8];
        #pragma unroll
        for (int i = 0; i < 4; ++i) {
            float rli = 1.0f / L[i][i];
            #pragma unroll
            for (int c = 0; c < 8; ++c) {
                float s = Bt[i][c];
                #pragma unroll
                for (int k = 0; k < 4; ++k) if (k < i) s -= L[i][k] * Y[k][c];
                Y[i][c] = s * rli;
            }
        }
        #pragma unroll
        for (int ii = 3; ii >= 0; --ii) {
            float rli = 1.0f / L[ii][ii];
            #pragma unroll
            for (int c = 0; c < 8; ++c) {
                float s = Y[ii][c];
                #pragma unroll
                for (int k = 0; k < 4; ++k) if (k > ii) s -= L[k][ii] * X[k][c];
                X[ii][c] = s * rli;
            }
        }
        // innovation + new mean: K = X^T
        float inn[4];
        #pragma unroll
        for (int a = 0; a < 4; ++a) inn[a] = zdet[a] - zm[a];
        float nm[8];
        #pragma unroll
        for (int i = 0; i < 8; ++i) {
            float s = pm[i];
            #pragma unroll
            for (int a = 0; a < 4; ++a) s += X[a][i] * inn[a];
            nm[i] = s;
        }
        // new cov = PC - K S K^T ;  K[i][a] = X[a][i]
        float KS[8][4];
        #pragma unroll
        for (int i = 0; i < 8; ++i)
            #pragma unroll
            for (int a = 0; a < 4; ++a) {
                float s = 0.f;
                #pragma unroll
                for (int b = 0; b < 4; ++b) s += X[b][i] * S[b][a];
                KS[i][a] = s;
            }
        // write outputs
        {
            float4* om = (float4*)(out_mean + (size_t)track * 8);
            om[0] = make_float4(nm[0], nm[1], nm[2], nm[3]);
            om[1] = make_float4(nm[4], nm[5], nm[6], nm[7]);
            float4* oc = (float4*)(out_cov + (size_t)track * 64);
            #pragma unroll
            for (int i = 0; i < 8; ++i) {
                float r[8];
                #pragma unroll
                for (int j = 0; j < 8; ++j) {
                    float s = PC[i][j];
                    #pragma unroll
                    for (int a = 0; a < 4; ++a) s -= KS[i][a] * X[a][j];
                    r[j] = s;
                }
                oc[i*2+0] = make_float4(r[0], r[1], r[2], r[3]);
                oc[i*2+1] = make_float4(r[4], r[5], r[6], r[7]);
            }
        }
        // gating features: W = L^{-1}, A4 = W^T W = S^{-1}
        float W[4][4];
        #pragma unroll
        for (int i = 0; i < 4; ++i)
            #pragma unroll
            for (int j = 0; j < 4; ++j) W[i][j] = 0.f;
        #pragma unroll
        for (int i = 0; i < 4; ++i) {
            float rli = 1.0f / L[i][i];
            W[i][i] = rli;
            #pragma unroll
            for (int j = 0; j < 4; ++j) if (j < i) {
                float s = 0.f;
                #pragma unroll
                for (int k = 0; k < 4; ++k) if (k >= j && k < i) s += L[i][k] * W[k][j];
                W[i][j] = -s * rli;
            }
        }
        float A4[4][4];
        #pragma unroll
        for (int i = 0; i < 4; ++i)
            #pragma unroll
            for (int j = 0; j < 4; ++j) {
                float s = 0.f;
                #pragma unroll
                for (int k = 0; k < 4; ++k) s += W[k][i] * W[k][j];
                A4[i][j] = s;
            }
        float bb[4];
        #pragma unroll
        for (int i = 0; i < 4; ++i) {
            float s = 0.f;
            #pragma unroll
            for (int j = 0; j < 4; ++j) s += A4[i][j] * zm[j];
            bb[i] = s;
        }
        float cterm = 0.f;
        #pragma unroll
        for (int i = 0; i < 4; ++i) cterm += bb[i] * zm[i];
        // t = [A_ij (x2 off-diag, 10), -2 b (4), mu^T A mu, 0]
        int idx = 0;
        #pragma unroll
        for (int i = 0; i < 4; ++i)
            #pragma unroll
            for (int j = i; j < 4; ++j)
                sTrk[tid][idx++] = (i == j) ? A4[i][j] : 2.0f * A4[i][j];
        #pragma unroll
        for (int i = 0; i < 4; ++i) sTrk[tid][10 + i] = -2.0f * bb[i];
        sTrk[tid][14] = cterm;
        sTrk[tid][15] = 0.0f;
    }
    __syncthreads();

    // =====================================================================
    // Phase 2: squared_maha[N,M] tile = T[16 x 16feat] @ Q^T[16feat x 16]
    // via 4 chained V_WMMA_F32_16X16X4_F32 per 16x16 tile. EXEC all ones.
    // =====================================================================
    const int lane = tid & 31;
    const int wave = tid >> 5;          // 0..7
    const int half = lane >> 4;         // K-half select per ISA 16x4 layout
    const int row  = lane & 15;         // M (A) / N (B) row within tile
    const int mtiles = M >> 4;          // 8 for M=128

    #pragma unroll
    for (int t = 0; t < 2; ++t) {
        const int tt = wave * 2 + t;    // track tile 0..15
        v2f afrag[4];
        #pragma unroll
        for (int kc = 0; kc < 4; ++kc) {
            afrag[kc].x = sTrk[tt * 16 + row][kc * 4 + half * 2 + 0];
            afrag[kc].y = sTrk[tt * 16 + row][kc * 4 + half * 2 + 1];
        }
        for (int mt = 0; mt < mtiles; ++mt) {
            v2f bfrag[4];
            #pragma unroll
            for (int kc = 0; kc < 4; ++kc) {
                bfrag[kc].x = sMeas[mt * 16 + row][kc * 4 + half * 2 + 0];
                bfrag[kc].y = sMeas[mt * 16 + row][kc * 4 + half * 2 + 1];
            }
            v8f acc = {0.f, 0.f, 0.f, 0.f, 0.f, 0.f, 0.f, 0.f};
            #pragma unroll
            for (int kc = 0; kc < 4; ++kc) {
                acc = __builtin_amdgcn_wmma_f32_16x16x4_f32(
                    false, afrag[kc], false, bfrag[kc],
                    (short)0, acc, false, false);
            }
            const int trkBase = blockIdx.x * NTRK_BLK + tt * 16;
            #pragma unroll
            for (int r = 0; r < 8; ++r) {
                out_maha[(size_t)(trkBase + r + half * 8) * M + mt * 16 + row] = acc[r];
            }
        }
    }
}

extern "C" void kernel_launch(void* const* d_in, const int* in_sizes, int n_in,
                              void* d_out, int out_size, void* d_ws, size_t ws_size,
                              hipStream_t stream) {
    const float* mean  = (const float*)d_in[0];
    const float* cov   = (const float*)d_in[1];
    const float* meas  = (const float*)d_in[2];
    const float* gmeas = (const float*)d_in[3];
    const float* Fm    = (const float*)d_in[4];
    const float* Hm    = (const float*)d_in[5];
    const float* Qm    = (const float*)d_in[6];
    const float* Rm    = (const float*)d_in[7];

    const int N = in_sizes[0] / 8;
    const int M = in_sizes[3] / 4;

    float* out_mean = (float*)d_out;
    float* out_cov  = out_mean + (size_t)N * 8;
    float* out_maha = out_cov  + (size_t)N * 64;

    dim3 grid((N + NTRK_BLK - 1) / NTRK_BLK), block(NTRK_BLK);
    hipLaunchKernelGGL(kalman_fused_kernel, grid, block, 0, stream,
                       mean, cov, meas, gmeas, Fm, Hm, Qm, Rm,
                       out_mean, out_cov, out_maha, N, M);
}